// CGCNN_74964359185021
// MI455X (gfx1250) — compile-verified
//
#include <hip/hip_runtime.h>

typedef __attribute__((ext_vector_type(2))) float v2f;
typedef __attribute__((ext_vector_type(8))) float v8f;

constexpr int NN  = 30000;   // nodes
constexpr int NE  = 480000;  // edges
constexpr int FIN = 64;
constexpr int FE  = 16;
constexpr int H   = 128;
constexpr int H2  = 256;
constexpr int NC  = 10;

#define V8ZERO {0.f,0.f,0.f,0.f,0.f,0.f,0.f,0.f}
#define USE_ASYNC_LDS 1

// ---------------------------------------------------------------------------
// C[M,N] = A[M,K] @ B[K,N] + bias[N], fp32 WMMA 16x16x4.
// Block = 8 waves sharing one 64-col n-group: the K x 64 B-strip is staged
// into LDS once (async global->LDS, ASYNCcnt-tracked), then each wave runs
// the WMMA K-loop on a 16-row m-tile, reading B fragments from LDS.
// Requires M%16==0, N%64==0, K%4==0 (all shapes here satisfy this).
// ---------------------------------------------------------------------------
__global__ void __launch_bounds__(256) k_gemm_bias(
    const float* __restrict__ A, const float* __restrict__ B,
    const float* __restrict__ bias, float* __restrict__ C,
    int M, int N, int K)
{
  extern __shared__ float Blds[];            // [K][64]
  const int tid = threadIdx.x;
  const int n0  = (int)blockIdx.y << 6;

  // ---- stage B strip (K x 64) into LDS: 16-byte groups, one per lane/iter
  const int groups = K * 16;                 // (K*64*4)/16
#if USE_ASYNC_LDS
  const unsigned ldsbase =
      (unsigned)(uintptr_t)(__attribute__((address_space(3))) float*)Blds;
  for (int i = tid; i < groups; i += 256) {
    const float* gp = B + (size_t)(i >> 4) * N + n0 + ((i & 15) << 2);
    const unsigned lo = ldsbase + (unsigned)i * 16u;
    asm volatile("global_load_async_to_lds_b128 %0, %1, off"
                 :: "v"(lo), "v"(gp) : "memory");
  }
  asm volatile("s_wait_asynccnt 0" ::: "memory");
#else
  for (int i = tid; i < groups; i += 256) {
    const float4 v = *(const float4*)(B + (size_t)(i >> 4) * N + n0 + ((i & 15) << 2));
    *(float4*)(Blds + (size_t)i * 4) = v;
  }
#endif
  __syncthreads();

  const int mtile = (int)blockIdx.x * 8 + (tid >> 5);
  if (mtile >= (M >> 4)) return;             // wave-uniform: EXEC stays all-1s

  const int lane = tid & 31;
  const int l16  = lane & 15;
  const int half = lane >> 4;
  const int m0 = mtile << 4;

  v8f acc0 = V8ZERO, acc1 = V8ZERO, acc2 = V8ZERO, acc3 = V8ZERO;

  const float* Arow = A + (size_t)(m0 + l16) * K;
  for (int k0 = 0; k0 < K; k0 += 4) {
    // A 16x4 f32 fragment: lanes 0-15 = M rows; comp j + half -> K = 2*half+j
    v2f a;
    a.x = Arow[k0 + 2 * half + 0];
    a.y = Arow[k0 + 2 * half + 1];
    // B 4x16 f32 fragments from LDS: lane l16 = column, rows k0+2*half(+1)
    const float* Bp = Blds + (size_t)(k0 + 2 * half) * 64 + l16;
    v2f b0, b1, b2, b3;
    b0.x = Bp[0];  b0.y = Bp[64];
    b1.x = Bp[16]; b1.y = Bp[80];
    b2.x = Bp[32]; b2.y = Bp[96];
    b3.x = Bp[48]; b3.y = Bp[112];
    acc0 = __builtin_amdgcn_wmma_f32_16x16x4_f32(false, a, false, b0, (short)0, acc0, false, false);
    acc1 = __builtin_amdgcn_wmma_f32_16x16x4_f32(false, a, false, b1, (short)0, acc1, false, false);
    acc2 = __builtin_amdgcn_wmma_f32_16x16x4_f32(false, a, false, b2, (short)0, acc2, false, false);
    acc3 = __builtin_amdgcn_wmma_f32_16x16x4_f32(false, a, false, b3, (short)0, acc3, false, false);
  }

  // C/D 16x16 f32 layout: VGPR r -> M = m0 + r + 8*half, N = n0 + l16 (+16j)
  const float bv0 = bias[n0 + l16];
  const float bv1 = bias[n0 + 16 + l16];
  const float bv2 = bias[n0 + 32 + l16];
  const float bv3 = bias[n0 + 48 + l16];
#pragma unroll
  for (int r = 0; r < 8; ++r) {
    float* Crow = C + (size_t)(m0 + r + 8 * half) * N + n0 + l16;
    Crow[0]  = acc0[r] + bv0;
    Crow[16] = acc1[r] + bv1;
    Crow[32] = acc2[r] + bv2;
    Crow[48] = acc3[r] + bv3;
  }
}

// ---------------------------------------------------------------------------
// Row-wise LayerNorm + ReLU. One wave per row, shuffle reductions (wave32).
// ---------------------------------------------------------------------------
template <int W>
__global__ void __launch_bounds__(256) k_ln_relu(
    const float* __restrict__ x, const float* __restrict__ g,
    const float* __restrict__ b, float* __restrict__ y, int rows)
{
  const int row = blockIdx.x * (blockDim.x >> 5) + (int)(threadIdx.x >> 5);
  if (row >= rows) return;
  const int lane = threadIdx.x & 31;
  constexpr int PER = W / 32;
  const float* xr = x + (size_t)row * W;
  float v[PER];
  float s = 0.f;
#pragma unroll
  for (int c = 0; c < PER; ++c) { v[c] = xr[c * 32 + lane]; s += v[c]; }
#pragma unroll
  for (int o = 16; o > 0; o >>= 1) s += __shfl_xor(s, o, 32);
  const float mean = s * (1.0f / W);
  float vs = 0.f;
#pragma unroll
  for (int c = 0; c < PER; ++c) { float d = v[c] - mean; vs += d * d; }
#pragma unroll
  for (int o = 16; o > 0; o >>= 1) vs += __shfl_xor(vs, o, 32);
  const float inv = rsqrtf(vs * (1.0f / W) + 1e-5f);
  float* yr = y + (size_t)row * W;
#pragma unroll
  for (int c = 0; c < PER; ++c) {
    const int f = c * 32 + lane;
    const float val = (v[c] - mean) * inv * g[f] + b[f];
    yr[f] = fmaxf(val, 0.f);
  }
}

// ---------------------------------------------------------------------------
__global__ void k_init_seg(float* __restrict__ nmax, float* __restrict__ nsum,
                           float* __restrict__ nmsm, int n)
{
  const int i = blockIdx.x * blockDim.x + threadIdx.x;
  if (i < n) { nmax[i] = -3.0e38f; nsum[i] = 0.f; nmsm[i] = 0.f; }
}

// Pass 1: msg = relu(z[src]+ea)+eps ; atomicMax(nmax[dst], msg*t)
__global__ void __launch_bounds__(256) k_edge_max(
    const float* __restrict__ z, const float* __restrict__ ea,
    const int* __restrict__ src, const int* __restrict__ dst,
    const float* __restrict__ tptr, float* __restrict__ nmax, int E)
{
  const int e = blockIdx.x * (blockDim.x >> 5) + (int)(threadIdx.x >> 5);
  if (e >= E) return;
  const int lane = threadIdx.x & 31;
  // prefetch the gathered node row (and edge row) ~64 edges ahead
  if (e + 64 < E) {
    __builtin_prefetch(z + (size_t)src[e + 64] * H, 0, 3);
    __builtin_prefetch(ea + (size_t)(e + 64) * H, 0, 3);
  }
  const float t = tptr[0];
  const int s = src[e], d = dst[e];
  const float4 zv = *(const float4*)(z + (size_t)s * H + lane * 4);
  const float4 ev = *(const float4*)(ea + (size_t)e * H + lane * 4);
  const float m0 = fmaxf(zv.x + ev.x, 0.f) + 1e-7f;
  const float m1 = fmaxf(zv.y + ev.y, 0.f) + 1e-7f;
  const float m2 = fmaxf(zv.z + ev.z, 0.f) + 1e-7f;
  const float m3 = fmaxf(zv.w + ev.w, 0.f) + 1e-7f;
  float* np = nmax + (size_t)d * H + lane * 4;
  atomicMax(np + 0, m0 * t);
  atomicMax(np + 1, m1 * t);
  atomicMax(np + 2, m2 * t);
  atomicMax(np + 3, m3 * t);
}

// Pass 2: e = exp(msg*t - max) ; nsum += e ; nmsm += msg*e
__global__ void __launch_bounds__(256) k_edge_sum(
    const float* __restrict__ z, const float* __restrict__ ea,
    const int* __restrict__ src, const int* __restrict__ dst,
    const float* __restrict__ tptr, const float* __restrict__ nmax,
    float* __restrict__ nsum, float* __restrict__ nmsm, int E)
{
  const int e = blockIdx.x * (blockDim.x >> 5) + (int)(threadIdx.x >> 5);
  if (e >= E) return;
  const int lane = threadIdx.x & 31;
  if (e + 64 < E) {
    __builtin_prefetch(z + (size_t)src[e + 64] * H, 0, 3);
    __builtin_prefetch(ea + (size_t)(e + 64) * H, 0, 3);
  }
  const float t = tptr[0];
  const int s = src[e], d = dst[e];
  const float4 zv = *(const float4*)(z + (size_t)s * H + lane * 4);
  const float4 ev = *(const float4*)(ea + (size_t)e * H + lane * 4);
  const float m0 = fmaxf(zv.x + ev.x, 0.f) + 1e-7f;
  const float m1 = fmaxf(zv.y + ev.y, 0.f) + 1e-7f;
  const float m2 = fmaxf(zv.z + ev.z, 0.f) + 1e-7f;
  const float m3 = fmaxf(zv.w + ev.w, 0.f) + 1e-7f;
  const float4 mx = *(const float4*)(nmax + (size_t)d * H + lane * 4);
  const float e0 = __expf(m0 * t - mx.x);
  const float e1 = __expf(m1 * t - mx.y);
  const float e2 = __expf(m2 * t - mx.z);
  const float e3 = __expf(m3 * t - mx.w);
  float* sp = nsum + (size_t)d * H + lane * 4;
  float* mp = nmsm + (size_t)d * H + lane * 4;
  atomicAdd(sp + 0, e0); atomicAdd(mp + 0, m0 * e0);
  atomicAdd(sp + 1, e1); atomicAdd(mp + 1, m1 * e1);
  atomicAdd(sp + 2, e2); atomicAdd(mp + 2, m2 * e2);
  atomicAdd(sp + 3, e3); atomicAdd(mp + 3, m3 * e3);
}

// zin = nmsm/(nsum + 1e-16) + xin
__global__ void k_combine(const float* __restrict__ nmsm, const float* __restrict__ nsum,
                          const float* __restrict__ xin, float* __restrict__ zin, int n)
{
  const int i = blockIdx.x * blockDim.x + threadIdx.x;
  if (i < n) zin[i] = nmsm[i] / (nsum[i] + 1e-16f) + xin[i];
}

// h = convout (+ h if addRes)
__global__ void k_update(const float* __restrict__ convout, float* __restrict__ h,
                         int n, int addRes)
{
  const int i = blockIdx.x * blockDim.x + threadIdx.x;
  if (i < n) h[i] = addRes ? (h[i] + convout[i]) : convout[i];
}

// out[n,c] = z[n,:] . lin_w[:,c] + lin_b[c]   (tiny: 30000x10)
__global__ void k_classifier(const float* __restrict__ z, const float* __restrict__ w,
                             const float* __restrict__ b, float* __restrict__ out, int n)
{
  const int idx = blockIdx.x * blockDim.x + threadIdx.x;
  if (idx >= n * NC) return;
  const int node = idx / NC;
  const int c = idx - node * NC;
  const float* zr = z + (size_t)node * H;
  float s = b[c];
#pragma unroll 8
  for (int f = 0; f < H; ++f) s += zr[f] * w[f * NC + c];
  out[idx] = s;
}

// ---------------------------------------------------------------------------
extern "C" void kernel_launch(void* const* d_in, const int* in_sizes, int n_in,
                              void* d_out, int out_size, void* d_ws, size_t ws_size,
                              hipStream_t stream)
{
  (void)in_sizes; (void)n_in; (void)out_size; (void)ws_size;
  const float* x      = (const float*)d_in[0];
  const float* eattr  = (const float*)d_in[1];
  const float* nenc_w = (const float*)d_in[2];
  const float* nenc_b = (const float*)d_in[3];
  const float* eenc_w = (const float*)d_in[4];
  const float* eenc_b = (const float*)d_in[5];
  const float* mlp1_w = (const float*)d_in[6];
  const float* mlp1_b = (const float*)d_in[7];
  const float* lng    = (const float*)d_in[8];
  const float* lnb    = (const float*)d_in[9];
  const float* mlp2_w = (const float*)d_in[10];
  const float* mlp2_b = (const float*)d_in[11];
  const float* tparam = (const float*)d_in[12];
  const float* norm_g = (const float*)d_in[13];
  const float* norm_b = (const float*)d_in[14];
  const float* lin_w  = (const float*)d_in[15];
  const float* lin_b  = (const float*)d_in[16];
  const int*   eidx   = (const int*)d_in[17];
  const int* src = eidx;        // edge_index[0]
  const int* dst = eidx + NE;   // edge_index[1]

  char* wsp = (char*)d_ws;
  auto alloc = [&](size_t bytes) -> float* {
    char* p = wsp;
    wsp += (bytes + 255) & ~(size_t)255;
    return (float*)p;
  };
  float* ea   = alloc((size_t)NE * H * 4);   // encoded edge attrs [E,128]
  float* h    = alloc((size_t)NN * H * 4);   // node state
  float* z    = alloc((size_t)NN * H * 4);   // pre-normed conv input
  float* zin  = alloc((size_t)NN * H * 4);   // agg + x
  float* nmax = alloc((size_t)NN * H * 4);   // segment max
  float* nsum = alloc((size_t)NN * H * 4);   // sum of exp
  float* nmsm = alloc((size_t)NN * H * 4);   // sum of msg*exp
  float* mid  = alloc((size_t)NN * H2 * 4);  // MLP hidden [N,256]
  float* cout = alloc((size_t)NN * H * 4);   // conv output

  auto gemm = [&](const float* A, const float* B, const float* bias, float* C,
                  int M, int N, int K) {
    dim3 grid(((M >> 4) + 7) / 8, N >> 6);
    const size_t shbytes = (size_t)K * 64 * 4;  // B strip K x 64 (<= 64KB)
    k_gemm_bias<<<grid, 256, shbytes, stream>>>(A, B, bias, C, M, N, K);
  };

  // encoders
  gemm(eattr, eenc_w, eenc_b, ea, NE, H, FE);
  gemm(x, nenc_w, nenc_b, h, NN, H, FIN);

  const int elemN = NN * H;
  const int eb = (elemN + 255) / 256;
  for (int i = 0; i < 3; ++i) {
    const float* zsrc;
    if (i == 0) {
      zsrc = h;  // first layer convolves raw encoded features
    } else {
      k_ln_relu<H><<<(NN + 7) / 8, 256, 0, stream>>>(h, norm_g + i * H, norm_b + i * H, z, NN);
      zsrc = z;
    }
    k_init_seg<<<eb, 256, 0, stream>>>(nmax, nsum, nmsm, elemN);
    k_edge_max<<<NE / 8, 256, 0, stream>>>(zsrc, ea, src, dst, tparam + i, nmax, NE);
    k_edge_sum<<<NE / 8, 256, 0, stream>>>(zsrc, ea, src, dst, tparam + i, nmax, nsum, nmsm, NE);
    k_combine<<<eb, 256, 0, stream>>>(nmsm, nsum, zsrc, zin, elemN);
    gemm(zin, mlp1_w + (size_t)i * H * H2, mlp1_b + i * H2, mid, NN, H2, H);
    k_ln_relu<H2><<<(NN + 7) / 8, 256, 0, stream>>>(mid, lng + i * H2, lnb + i * H2, mid, NN);
    gemm(mid, mlp2_w + (size_t)i * H2 * H, mlp2_b + i * H, cout, NN, H, H2);
    k_update<<<eb, 256, 0, stream>>>(cout, h, elemN, i > 0 ? 1 : 0);
  }

  // final norm/act (layer-0 params) + classifier
  k_ln_relu<H><<<(NN + 7) / 8, 256, 0, stream>>>(h, norm_g, norm_b, z, NN);
  k_classifier<<<(NN * NC + 255) / 256, 256, 0, stream>>>(z, lin_w, lin_b, (float*)d_out, NN);
}